// SpatialNonIntersectionAxiom_40570261078453
// MI455X (gfx1250) — compile-verified
//
#include <hip/hip_runtime.h>

#define EPSILON_C 0.001f
#define PROX_C    0.15f

typedef __attribute__((ext_vector_type(2))) float v2f;
typedef __attribute__((ext_vector_type(8))) float v8f;

// ---------------------------------------------------------------------------
// Workspace layout (E = edge count), float-indexed:
//   [0*E..1*E)  dx        [1*E..2*E)  dy
//   [2*E..3*E)  a1x       [3*E..4*E)  a1y
//   [4*E..5*E)  da1       [5*E..6*E)  a (= max(d.d,1e-12))
//   [6*E..7*E)  half      [7*E..8*E)  mid_sq
//   [8*E..9*E)  p_sq      [9*E..10*E) src (int)
//   [10*E..11*E) dst (int)
//   [11*E]      sum accumulator (float)
//   [11*E+1]    cand count accumulator (uint)
// ---------------------------------------------------------------------------

__global__ void zero_acc(float* ws, int E) {
  ws[(size_t)11 * E] = 0.0f;
  ((unsigned int*)ws)[(size_t)11 * E + 1] = 0u;
}

__global__ void edge_precompute(const float* __restrict__ pos,
                                const int* __restrict__ edge_index,
                                float* __restrict__ ws, int E) {
  int e = blockIdx.x * blockDim.x + threadIdx.x;
  if (e >= E) return;
  int s  = edge_index[e];
  int dn = edge_index[E + e];
  float a1x = pos[2 * s], a1y = pos[2 * s + 1];
  float a2x = pos[2 * dn], a2y = pos[2 * dn + 1];
  float dx = a2x - a1x, dy = a2y - a1y;
  float dd = dx * dx + dy * dy;
  float midx = 0.5f * (a1x + a2x), midy = 0.5f * (a1y + a2y);
  ws[e]          = dx;
  ws[E + e]      = dy;
  ws[2 * E + e]  = a1x;
  ws[3 * E + e]  = a1y;
  ws[4 * E + e]  = dx * a1x + dy * a1y;                        // da1
  ws[5 * E + e]  = fmaxf(dd, 1e-12f);                          // a
  ws[6 * E + e]  = 0.5f * __builtin_amdgcn_sqrtf(fmaxf(dd, 0.0f)); // half
  ws[7 * E + e]  = midx * midx + midy * midy;                  // mid_sq
  ws[8 * E + e]  = a1x * a1x + a1y * a1y;                      // p_sq
  ((int*)ws)[9 * E + e]  = s;
  ((int*)ws)[10 * E + e] = dn;
}

// One wave32 per 16x16 tile of the ExE pair matrix; 8 waves per block.
// Tiles strictly below the diagonal (tj < ti) are identically zero
// (cand requires j > i) -> store zeros, skip all math.
// For compute tiles, WMMA 16x16x4 F32 produces the four rank-2 Gram tiles:
//   b = d_i.d_j, P = d_i.a1_j, Q = a1_i.d_j, R = a1_i.a1_j
// mid_i.mid_j is reconstructed as R + 0.5*(P+Q) + 0.25*b.
__global__ __launch_bounds__(256) void pair_tiles(
    const float* __restrict__ ws,
    float* __restrict__ out_mask,
    float* __restrict__ out_pp,
    float* __restrict__ acc_sum,
    unsigned int* __restrict__ acc_cnt,
    int E, int totalTiles) {
  const int lane = threadIdx.x & 31;
  const int wave = threadIdx.x >> 5;
  const int tilesPerRow = E >> 4;
  const int tile = blockIdx.x * 8 + wave;
  const bool active = tile < totalTiles;

  const int ti = active ? (tile / tilesPerRow) : 0;
  const int tj = active ? (tile - ti * tilesPerRow) : 0;
  const int i0 = ti * 16, j0 = tj * 16;
  const bool compute = active && (tj >= ti);

  const float* dxv  = ws;
  const float* dyv  = ws + E;
  const float* a1xv = ws + 2 * E;
  const float* a1yv = ws + 3 * E;
  const float* da1v = ws + 4 * E;
  const float* av   = ws + 5 * E;
  const float* hv   = ws + 6 * E;
  const float* mqv  = ws + 7 * E;
  const float* pqv  = ws + 8 * E;
  const int*   srcv = (const int*)(ws + 9 * E);
  const int*   dstv = (const int*)(ws + 10 * E);

  // Per-wave staging of row (0..15) and column (16..31) scalars.
  // Stride 9 dwords to avoid LDS bank conflicts on the column reads.
  __shared__ float lds[8][32][9];

  if (compute) {
    int idx = (lane < 16) ? (i0 + lane) : (j0 + (lane - 16));
    lds[wave][lane][0] = da1v[idx];
    lds[wave][lane][1] = av[idx];
    lds[wave][lane][2] = hv[idx];
    lds[wave][lane][3] = mqv[idx];
    lds[wave][lane][4] = pqv[idx];
    ((int*)lds[wave][lane])[5] = srcv[idx];
    ((int*)lds[wave][lane])[6] = dstv[idx];
  }
  __syncthreads();

  if (!active) return;

  const int n  = lane & 15;
  const int hi = lane >> 4;
  const int j  = j0 + n;

  if (!compute) {
    // Whole tile lies strictly below the diagonal: outputs are zero.
#pragma unroll
    for (int k = 0; k < 8; ++k) {
      const int i = i0 + k + hi * 8;
      const size_t idx = (size_t)i * (size_t)E + (size_t)j;
      out_pp[idx]   = 0.0f;
      out_mask[idx] = 0.0f;
    }
    return;
  }

  // WMMA operands. 16x4 f32 A: lanes 0-15 carry K=0 (V0) / K=1 (V1) for
  // M = lane; lanes 16-31 carry K=2/3 which we zero-pad. 4x16 f32 B mirrors
  // this: lanes 0-15 carry K=0/1 rows for N = lane; upper half zero.
  v2f Ad = {0.0f, 0.0f}, Aa = {0.0f, 0.0f};
  v2f Bd = {0.0f, 0.0f}, Ba = {0.0f, 0.0f};
  if (lane < 16) {
    int i = i0 + lane;
    Ad.x = dxv[i];  Ad.y = dyv[i];
    Aa.x = a1xv[i]; Aa.y = a1yv[i];
    int jj = j0 + lane;
    Bd.x = dxv[jj];  Bd.y = dyv[jj];
    Ba.x = a1xv[jj]; Ba.y = a1yv[jj];
  }

  v8f zero = {};
  // (neg_a, A, neg_b, B, c_mod, C, reuse_a, reuse_b)
  v8f Bt = __builtin_amdgcn_wmma_f32_16x16x4_f32(false, Ad, false, Bd, (short)0, zero, false, false);
  v8f Pt = __builtin_amdgcn_wmma_f32_16x16x4_f32(false, Ad, false, Ba, (short)0, zero, false, false);
  v8f Qt = __builtin_amdgcn_wmma_f32_16x16x4_f32(false, Aa, false, Bd, (short)0, zero, false, false);
  v8f Rt = __builtin_amdgcn_wmma_f32_16x16x4_f32(false, Aa, false, Ba, (short)0, zero, false, false);

  // C/D layout: element (M = k + 8*(lane>=16), N = lane&15) lives in VGPR k.
  const float da1_j  = lds[wave][16 + n][0];
  const float a_j    = lds[wave][16 + n][1];
  const float half_j = lds[wave][16 + n][2];
  const float msq_j  = lds[wave][16 + n][3];
  const float psq_j  = lds[wave][16 + n][4];
  const int   src_j  = ((int*)lds[wave][16 + n])[5];
  const int   dst_j  = ((int*)lds[wave][16 + n])[6];
  const float rcp_aj = __builtin_amdgcn_rcpf(a_j);

  float local_sum = 0.0f;
  unsigned int local_cnt = 0u;

#pragma unroll
  for (int k = 0; k < 8; ++k) {
    const int m = k + hi * 8;
    const int i = i0 + m;
    const float b = Bt[k], P = Pt[k], Q = Qt[k], R = Rt[k];

    const float da1_i  = lds[wave][m][0];
    const float a_i    = lds[wave][m][1];
    const float half_i = lds[wave][m][2];
    const float msq_i  = lds[wave][m][3];
    const float psq_i  = lds[wave][m][4];
    const int   src_i  = ((int*)lds[wave][m])[5];
    const int   dst_i  = ((int*)lds[wave][m])[6];

    // candidate mask: midd < reach  <=>  mid2 < reach^2 (reach > 0)
    const float mid2 = msq_i + msq_j - 2.0f * R - (P + Q) - 0.5f * b;
    const float reach = half_i + half_j + PROX_C;
    const bool share = (src_i == src_j) | (src_i == dst_j) |
                       (dst_i == src_j) | (dst_i == dst_j);
    const bool cand = (j > i) & (!share) & (mid2 < reach * reach);

    // segment-segment closest-point parameters (fast reciprocals)
    const float c = da1_i - P;      // c_ij = da1_i - d_i.a1_j
    const float f = Q - da1_j;      // f_ij = a1_i.d_j - da1_j
    const float denom = fmaxf(a_i * a_j - b * b, 1e-12f);
    const float rcp_den = __builtin_amdgcn_rcpf(denom);
    float s = fminf(fmaxf((b * f - c * a_j) * rcp_den, 0.0f), 1.0f);
    float t = fminf(fmaxf((b * s + f) * rcp_aj, 0.0f), 1.0f);
    const float rcp_ai = __builtin_amdgcn_rcpf(a_i);
    s = fminf(fmaxf((b * t - c) * rcp_ai, 0.0f), 1.0f);

    const float r2 = psq_i + psq_j - 2.0f * R;
    const float dist2 = r2 + s * (s * a_i + 2.0f * c)
                           + t * (t * a_j - 2.0f * f)
                           - 2.0f * s * t * b;
    const float dist = __builtin_amdgcn_sqrtf(fmaxf(dist2, 1e-24f));
    const float pp = cand ? fmaxf(EPSILON_C - dist, 0.0f) : 0.0f;

    const size_t idx = (size_t)i * (size_t)E + (size_t)j;
    out_pp[idx]   = pp;
    out_mask[idx] = (pp > 0.0f) ? 1.0f : 0.0f;
    local_sum += pp;
    local_cnt += cand ? 1u : 0u;
  }

  // wave32 reduction, one atomic per wave
  for (int off = 16; off > 0; off >>= 1) {
    local_sum += __shfl_xor(local_sum, off, 32);
    local_cnt += __shfl_xor(local_cnt, off, 32);
  }
  if (lane == 0) {
    atomicAdd(acc_sum, local_sum);
    atomicAdd(acc_cnt, local_cnt);
  }
}

__global__ void finalize(const float* acc_sum, const unsigned int* acc_cnt,
                         float* out) {
  unsigned int c = *acc_cnt;
  float npairs = (float)(c ? c : 1u);
  out[0] = (*acc_sum) / npairs;
}

extern "C" void kernel_launch(void* const* d_in, const int* in_sizes, int n_in,
                              void* d_out, int out_size, void* d_ws, size_t ws_size,
                              hipStream_t stream) {
  const float* pos = (const float*)d_in[0];
  // d_in[1] (adjacency) is unused by the reference computation.
  const int* edge_index = (const int*)d_in[2];
  const int E = in_sizes[2] / 2;  // edge_index is (2, E)

  float* ws  = (float*)d_ws;
  float* out = (float*)d_out;
  const size_t EE = (size_t)E * (size_t)E;
  float* out_mask = out + 1;
  float* out_pp   = out + 1 + EE;
  float* acc_sum  = ws + (size_t)11 * E;
  unsigned int* acc_cnt = (unsigned int*)(ws + (size_t)11 * E + 1);

  zero_acc<<<1, 1, 0, stream>>>(ws, E);
  edge_precompute<<<(E + 255) / 256, 256, 0, stream>>>(pos, edge_index, ws, E);

  const int tilesPerRow = E / 16;
  const int totalTiles = tilesPerRow * tilesPerRow;   // 65536 for E=4096
  const int blocks = (totalTiles + 7) / 8;            // 8 waves (tiles) / block
  pair_tiles<<<blocks, 256, 0, stream>>>(ws, out_mask, out_pp, acc_sum, acc_cnt,
                                         E, totalTiles);
  finalize<<<1, 1, 0, stream>>>(acc_sum, acc_cnt, out);
}